// ConditionalControlModule_13915694039322
// MI455X (gfx1250) — compile-verified
//
#include <hip/hip_runtime.h>
#include <hip/hip_bf16.h>

#define HSV_BINS 32
#define LBP_BINS 256
#define IMG      512
#define TILE     32
#define HALO     34       // TILE + 2 halo
#define NFEAT    2912     // 2048 high + 96 hsv-hist + 768 lbp-hist
#define NOUT     512
#define NBATCH   4
#define NT       8

typedef float v2f __attribute__((ext_vector_type(2)));
typedef float v8f __attribute__((ext_vector_type(8)));

// ---------------------------------------------------------------------------
// Fused HSV + histogram + LBP kernel. One pass over the 100 MB of frames.
// Grid: (16, 16, 32 frames), 256 threads (8 wave32 waves) per block.
// ---------------------------------------------------------------------------
__global__ __launch_bounds__(256) void hsv_lbp_hist_kernel(
    const float* __restrict__ seq,          // [32][3][512][512]
    unsigned int* __restrict__ hsvCnt,      // [32][3][32]
    unsigned int* __restrict__ lbpCnt)      // [32][3][256]
{
    __shared__ float        s_hsv[3 * HALO * HALO];                 // ~13.9 KB
    __shared__ unsigned int s_hist[3 * (HSV_BINS + LBP_BINS)];      // 864 bins

    const int tid = threadIdx.x;
    const int f   = blockIdx.z;
    const int ty  = blockIdx.y * TILE;
    const int tx  = blockIdx.x * TILE;
    const float* base = seq + (size_t)f * 3 * IMG * IMG;

    for (int i = tid; i < 3 * (HSV_BINS + LBP_BINS); i += 256) s_hist[i] = 0u;

    // Load RGB halo (reflect padding) and convert to HSV in LDS.
    for (int i = tid; i < HALO * HALO; i += 256) {
        int py = i / HALO, px = i % HALO;
        int y = ty + py - 1;
        int x = tx + px - 1;
        y = (y < 0) ? -y : ((y >= IMG) ? (2 * IMG - 2 - y) : y);   // jnp 'reflect'
        x = (x < 0) ? -x : ((x >= IMG) ? (2 * IMG - 2 - x) : x);
        const int o = y * IMG + x;
        float r = fminf(fmaxf(base[o],                 0.f), 1.f);
        float g = fminf(fmaxf(base[IMG * IMG + o],     0.f), 1.f);
        float b = fminf(fmaxf(base[2 * IMG * IMG + o], 0.f), 1.f);

        float maxc  = fmaxf(r, fmaxf(g, b));
        float minc  = fminf(r, fminf(g, b));
        float delta = maxc - minc;
        bool  mask  = delta > 1e-6f;
        float safe  = mask ? delta : 1.0f;
        float hue   = 0.f;
        if (mask && maxc == r) {                 // jnp.mod semantics (sign of divisor)
            float h = fmodf((g - b) / safe, 6.f);
            if (h < 0.f) h += 6.f;
            hue = h;
        }
        if (mask && maxc == g) hue = (b - r) / safe + 2.f;   // later 'where' wins
        if (mask && maxc == b) hue = (r - g) / safe + 4.f;
        hue *= (1.f / 6.f);
        float sat = (maxc > 1e-6f) ? delta / fmaxf(maxc, 1e-6f) : 0.f;

        s_hsv[0 * HALO * HALO + i] = hue;
        s_hsv[1 * HALO * HALO + i] = sat;
        s_hsv[2 * HALO * HALO + i] = maxc;
    }
    __syncthreads();

    // Histogram + LBP for the 32x32 interior (4 pixels/thread).
    for (int p = tid; p < TILE * TILE; p += 256) {
        const int cy = p / TILE + 1, cx = p % TILE + 1;
        #pragma unroll
        for (int c = 0; c < 3; ++c) {
            const float* hp = &s_hsv[c * HALO * HALO];
            const float ctr = hp[cy * HALO + cx];

            int bin = (int)floorf(ctr * (float)HSV_BINS);
            bin = min(max(bin, 0), HSV_BINS - 1);
            atomicAdd(&s_hist[c * HSV_BINS + bin], 1u);

            // bit order per reference offsets table
            int code = 0;
            code |= (hp[(cy - 1) * HALO + (cx - 1)] >= ctr) ?   1 : 0;
            code |= (hp[(cy - 1) * HALO +  cx     ] >= ctr) ?   2 : 0;
            code |= (hp[(cy - 1) * HALO + (cx + 1)] >= ctr) ?   4 : 0;
            code |= (hp[ cy      * HALO + (cx + 1)] >= ctr) ?   8 : 0;
            code |= (hp[(cy + 1) * HALO + (cx + 1)] >= ctr) ?  16 : 0;
            code |= (hp[(cy + 1) * HALO +  cx     ] >= ctr) ?  32 : 0;
            code |= (hp[(cy + 1) * HALO + (cx - 1)] >= ctr) ?  64 : 0;
            code |= (hp[ cy      * HALO + (cx - 1)] >= ctr) ? 128 : 0;
            atomicAdd(&s_hist[3 * HSV_BINS + c * LBP_BINS + code], 1u);
        }
    }
    __syncthreads();

    // Flush per-block partials to global histograms.
    for (int i = tid; i < 3 * (HSV_BINS + LBP_BINS); i += 256) {
        unsigned int v = s_hist[i];
        if (v) {
            if (i < 3 * HSV_BINS) atomicAdd(&hsvCnt[f * (3 * HSV_BINS) + i], v);
            else atomicAdd(&lbpCnt[f * (3 * LBP_BINS) + (i - 3 * HSV_BINS)], v);
        }
    }
}

// ---------------------------------------------------------------------------
// Fuse high + normalized histograms, mean over T=8 -> aggregated [4][2912].
// Normalization denominator is exactly H*W = 262144 (every pixel bins once).
// ---------------------------------------------------------------------------
__global__ __launch_bounds__(256) void aggregate_kernel(
    const float* __restrict__ high,              // [32][2048]
    const unsigned int* __restrict__ hsvCnt,     // [32][96]
    const unsigned int* __restrict__ lbpCnt,     // [32][768]
    float* __restrict__ agg)                     // [4][2912]
{
    const int j = blockIdx.x * 256 + threadIdx.x;
    if (j >= NBATCH * NFEAT) return;
    const int bb  = j / NFEAT;
    const int col = j % NFEAT;
    float s = 0.f;
    if (col < 2048) {
        #pragma unroll
        for (int t = 0; t < NT; ++t) s += high[(size_t)(bb * NT + t) * 2048 + col];
        s *= (1.f / NT);
    } else if (col < 2048 + 96) {
        const int jj = col - 2048;
        #pragma unroll
        for (int t = 0; t < NT; ++t) s += (float)hsvCnt[(bb * NT + t) * 96 + jj];
        s *= (1.f / (NT * 262144.f));
    } else {
        const int jj = col - 2144;
        #pragma unroll
        for (int t = 0; t < NT; ++t) s += (float)lbpCnt[(bb * NT + t) * 768 + jj];
        s *= (1.f / (NT * 262144.f));
    }
    agg[j] = s;
}

// ---------------------------------------------------------------------------
// z = agg @ W^T + b, leaky-ReLU(0.2), via V_WMMA_F32_16X16X4_F32.
// A (M=4 padded to 16) staged in LDS with an explicit zero row so the K-loop
// has NO conditional loads: per step it is one ds_load_b64 (A), one
// global_load_b64 (B), one v_wmma. 8 waves x 4 N-tiles of 16 = N=512.
// Also emits the one-hot c_cls block. Single 256-thread block.
// ---------------------------------------------------------------------------
__global__ __launch_bounds__(256) void head_wmma_kernel(
    const float* __restrict__ agg,      // [4][2912]
    const float* __restrict__ Wmat,     // [512][2912]
    const float* __restrict__ bias,     // [512]
    const int* __restrict__ env,        // [4]
    const int* __restrict__ season,     // [4]
    float* __restrict__ out)            // [4*512] c_feat | [4*8] c_cls
{
    // 5 rows: rows 0-3 = agg, row 4 = zeros (A-pad source for M=4..15)
    __shared__ float s_agg[5 * NFEAT];                 // ~58 KB

    const int tid   = threadIdx.x;
    const int lane  = tid & 31;
    const int wave  = tid >> 5;
    const int l16   = lane & 15;
    const int khalf = (lane >> 4) << 1;                // 0 for lanes 0-15, 2 for 16-31

    for (int i = tid; i < 5 * NFEAT; i += 256)
        s_agg[i] = (i < NBATCH * NFEAT) ? agg[i] : 0.f;
    __syncthreads();

    // Fixed per-lane A row pointer: real row for M<4, zero row otherwise.
    const float* arow = &s_agg[(size_t)((l16 < NBATCH) ? l16 : NBATCH) * NFEAT + khalf];

    for (int ntile = wave; ntile < NOUT / 16; ntile += 8) {
        const int n = ntile * 16 + l16;
        const float* wrow = Wmat + (size_t)n * NFEAT + khalf;   // B[k][n] = W[n][k]
        v8f acc = {};
        #pragma unroll 4
        for (int k = 0; k < NFEAT; k += 4) {
            const v2f a  = *(const v2f*)(arow + k);             // ds_load_b64
            const v2f bm = *(const v2f*)(wrow + k);             // global_load_b64
            acc = __builtin_amdgcn_wmma_f32_16x16x4_f32(
                false, a, false, bm, (short)0, acc, false, false);
        }
        if (lane < 16) {                                // D: VGPR v = row M=v, N=lane
            const float bv = bias[n];
            #pragma unroll
            for (int m = 0; m < NBATCH; ++m) {
                float z = acc[m] + bv;
                out[m * NOUT + n] = (z >= 0.f) ? z : 0.2f * z;
            }
        }
    }

    if (tid < NBATCH * 8) {                             // c_cls one-hots
        const int row = tid >> 3, col = tid & 7;
        float v = (col < 4) ? ((env[row] == col) ? 1.f : 0.f)
                            : ((season[row] == col - 4) ? 1.f : 0.f);
        out[NBATCH * NOUT + tid] = v;
    }
}

// ---------------------------------------------------------------------------
extern "C" void kernel_launch(void* const* d_in, const int* in_sizes, int n_in,
                              void* d_out, int out_size, void* d_ws, size_t ws_size,
                              hipStream_t stream) {
    const float* seq    = (const float*)d_in[0];   // (4,8,3,512,512) f32
    const float* high   = (const float*)d_in[1];   // (32,2048) f32
    const float* Wmat   = (const float*)d_in[2];   // (512,2912) f32
    const float* bias   = (const float*)d_in[3];   // (512,) f32
    const int*   env    = (const int*)d_in[4];     // (4,) i32
    const int*   season = (const int*)d_in[5];     // (4,) i32
    float* out = (float*)d_out;                    // 4*512 + 4*8 f32

    // workspace: hsv counts | lbp counts | aggregated
    unsigned int* hsvCnt = (unsigned int*)d_ws;                    // 32*96 u32
    unsigned int* lbpCnt = hsvCnt + 32 * 96;                       // 32*768 u32
    float*        agg    = (float*)(lbpCnt + 32 * 768);            // 4*2912 f32

    hipMemsetAsync(d_ws, 0, (size_t)(32 * 96 + 32 * 768) * sizeof(unsigned int),
                   stream);

    dim3 g1(IMG / TILE, IMG / TILE, 32);
    hsv_lbp_hist_kernel<<<g1, 256, 0, stream>>>(seq, hsvCnt, lbpCnt);

    aggregate_kernel<<<(NBATCH * NFEAT + 255) / 256, 256, 0, stream>>>(
        high, hsvCnt, lbpCnt, agg);

    head_wmma_kernel<<<1, 256, 0, stream>>>(agg, Wmat, bias, env, season, out);
}